// SAGENet_10307921511081
// MI455X (gfx1250) — compile-verified
//
#include <hip/hip_runtime.h>
#include <hip/hip_bf16.h>

typedef float v2f __attribute__((ext_vector_type(2)));
typedef float v8f __attribute__((ext_vector_type(8)));

// D = A(16x4 f32) * B(4x16 f32) + C(16x16 f32), wave32.
// A frag: lane l (l&15 = row M), VGPR0 = K=2*(l>>4), VGPR1 = K+1
// B frag: lane l (l&15 = col N), VGPR0 = K=2*(l>>4), VGPR1 = K+1
// C frag: VGPR r -> row M = r + 8*(l>>4), col N = l&15
#define WMMA_F32_TILE(a, b, c) \
  __builtin_amdgcn_wmma_f32_16x16x4_f32(false, (a), false, (b), (short)0, (c), false, false)

// Non-returning hardware fp32 atomic add (global_atomic_add_f32, STOREcnt path)
__device__ __forceinline__ void atomic_add_f32(float* p, float v) {
  unsafeAtomicAdd(p, v);
}

// ---------------------------------------------------------------------------
// h[N,64] = x[N,64] @ W[64,64] + bias[64]
// block = 128 threads (4 waves); wave w owns output cols [16w, 16w+16)
// grid.x = ceil(N/16) M-tiles
// ---------------------------------------------------------------------------
__global__ void lin_proj_wmma(const float* __restrict__ x,
                              const float* __restrict__ W,
                              const float* __restrict__ bias,
                              float* __restrict__ h, int n) {
  const int lane = threadIdx.x & 31;
  const int wave = threadIdx.x >> 5;
  const int m0   = blockIdx.x * 16;
  const int nb   = wave * 16;
  const int lid  = lane & 15;
  const int half = lane >> 4;

  int arow = m0 + lid;
  if (arow >= n) arow = n - 1;  // clamp loads; EXEC stays all-ones for WMMA

  v8f c = {0.f, 0.f, 0.f, 0.f, 0.f, 0.f, 0.f, 0.f};
  for (int k = 0; k < 64; k += 4) {
    const int kk = k + 2 * half;
    v2f a, b;
    a[0] = x[arow * 64 + kk];
    a[1] = x[arow * 64 + kk + 1];
    b[0] = W[kk * 64 + nb + lid];
    b[1] = W[(kk + 1) * 64 + nb + lid];
    c = WMMA_F32_TILE(a, b, c);
  }
  const float bv = bias[nb + lid];
#pragma unroll
  for (int r = 0; r < 8; ++r) {
    const int orow = m0 + r + 8 * half;
    if (orow < n) h[orow * 64 + nb + lid] = c[r] + bv;
  }
}

// ---------------------------------------------------------------------------
// deg[d] += 1 per incoming edge
// ---------------------------------------------------------------------------
__global__ void degree_kernel(const int* __restrict__ dst,
                              float* __restrict__ deg, int E) {
  int e = blockIdx.x * blockDim.x + threadIdx.x;
  if (e < E) atomic_add_f32(&deg[dst[e]], 1.0f);
}

// ---------------------------------------------------------------------------
// sum[dst[e], :F] += feat[src[e], :F]   (float4 gather, 4x fp32 atomics)
// One thread per (edge, 4-float chunk); chunks = F/4 must be a power of two.
// ---------------------------------------------------------------------------
__global__ void scatter_add(const int* __restrict__ src,
                            const int* __restrict__ dst,
                            const float* __restrict__ feat,
                            float* __restrict__ sum,
                            int E, int chunk_shift, int F) {
  const int idx    = blockIdx.x * blockDim.x + threadIdx.x;
  const int chunks = 1 << chunk_shift;
  const int e      = idx >> chunk_shift;
  if (e >= E) return;
  const int ch = idx & (chunks - 1);
  const int s  = src[e];
  const int d  = dst[e];
  const float4 v = *(const float4*)(feat + (size_t)s * F + ch * 4);
  float* p = sum + (size_t)d * F + ch * 4;
  atomic_add_f32(p + 0, v.x);
  atomic_add_f32(p + 1, v.y);
  atomic_add_f32(p + 2, v.z);
  atomic_add_f32(p + 3, v.w);
}

// ---------------------------------------------------------------------------
// h1[N,16] = tanh( (sum1/deg) @ W1_agg[64,16] + h @ W1_root[64,16] + b1[16] )
// One wave per 16-row tile (output is a single 16-col tile); 8 waves/block.
// ---------------------------------------------------------------------------
__global__ void combine1_wmma(const float* __restrict__ sum1,
                              const float* __restrict__ deg,
                              const float* __restrict__ h,
                              const float* __restrict__ Wa,
                              const float* __restrict__ Wr,
                              const float* __restrict__ b1,
                              float* __restrict__ h1, int n) {
  const int lane   = threadIdx.x & 31;
  const int wave   = threadIdx.x >> 5;
  const int tiles  = (n + 15) / 16;
  const int m_tile = blockIdx.x * 8 + wave;
  if (m_tile >= tiles) return;  // wave-uniform exit
  const int m0   = m_tile * 16;
  const int lid  = lane & 15;
  const int half = lane >> 4;

  int arow = m0 + lid;
  if (arow >= n) arow = n - 1;
  const float scale = 1.0f / fmaxf(deg[arow], 1.0f);

  v8f c = {0.f, 0.f, 0.f, 0.f, 0.f, 0.f, 0.f, 0.f};
  for (int k = 0; k < 64; k += 4) {
    const int kk = k + 2 * half;
    v2f aa, ar, ba, br;
    aa[0] = sum1[arow * 64 + kk] * scale;
    aa[1] = sum1[arow * 64 + kk + 1] * scale;
    ar[0] = h[arow * 64 + kk];
    ar[1] = h[arow * 64 + kk + 1];
    ba[0] = Wa[kk * 16 + lid];
    ba[1] = Wa[(kk + 1) * 16 + lid];
    br[0] = Wr[kk * 16 + lid];
    br[1] = Wr[(kk + 1) * 16 + lid];
    c = WMMA_F32_TILE(aa, ba, c);
    c = WMMA_F32_TILE(ar, br, c);
  }
  const float bv = b1[lid];
#pragma unroll
  for (int r = 0; r < 8; ++r) {
    const int orow = m0 + r + 8 * half;
    if (orow < n) h1[orow * 16 + lid] = tanhf(c[r] + bv);
  }
}

// ---------------------------------------------------------------------------
// out[N,64] = tanh( (sum2/deg) @ W2_agg[16,64] + h1 @ W2_root[16,64] + b2[64] )
// block = 128 (4 waves), wave w owns cols [16w,16w+16); grid.x = ceil(N/16)
// ---------------------------------------------------------------------------
__global__ void combine2_wmma(const float* __restrict__ sum2,
                              const float* __restrict__ deg,
                              const float* __restrict__ h1,
                              const float* __restrict__ Wa,
                              const float* __restrict__ Wr,
                              const float* __restrict__ b2,
                              float* __restrict__ out, int n) {
  const int lane = threadIdx.x & 31;
  const int wave = threadIdx.x >> 5;
  const int m0   = blockIdx.x * 16;
  const int nb   = wave * 16;
  const int lid  = lane & 15;
  const int half = lane >> 4;

  int arow = m0 + lid;
  if (arow >= n) arow = n - 1;
  const float scale = 1.0f / fmaxf(deg[arow], 1.0f);

  v8f c = {0.f, 0.f, 0.f, 0.f, 0.f, 0.f, 0.f, 0.f};
  for (int k = 0; k < 16; k += 4) {
    const int kk = k + 2 * half;
    v2f aa, ar, ba, br;
    aa[0] = sum2[arow * 16 + kk] * scale;
    aa[1] = sum2[arow * 16 + kk + 1] * scale;
    ar[0] = h1[arow * 16 + kk];
    ar[1] = h1[arow * 16 + kk + 1];
    ba[0] = Wa[kk * 64 + nb + lid];
    ba[1] = Wa[(kk + 1) * 64 + nb + lid];
    br[0] = Wr[kk * 64 + nb + lid];
    br[1] = Wr[(kk + 1) * 64 + nb + lid];
    c = WMMA_F32_TILE(aa, ba, c);
    c = WMMA_F32_TILE(ar, br, c);
  }
  const float bv = b2[nb + lid];
#pragma unroll
  for (int r = 0; r < 8; ++r) {
    const int orow = m0 + r + 8 * half;
    if (orow < n) out[orow * 64 + nb + lid] = tanhf(c[r] + bv);
  }
}

// ---------------------------------------------------------------------------
extern "C" void kernel_launch(void* const* d_in, const int* in_sizes, int n_in,
                              void* d_out, int out_size, void* d_ws, size_t ws_size,
                              hipStream_t stream) {
  const float* x      = (const float*)d_in[0];
  const int*   eidx   = (const int*)d_in[1];
  const float* W_lin  = (const float*)d_in[2];
  const float* b_lin  = (const float*)d_in[3];
  const float* W1_agg = (const float*)d_in[4];
  const float* W1_root= (const float*)d_in[5];
  const float* b1     = (const float*)d_in[6];
  const float* W2_agg = (const float*)d_in[7];
  const float* W2_root= (const float*)d_in[8];
  const float* b2     = (const float*)d_in[9];
  float* out = (float*)d_out;

  const int N = in_sizes[0] / 64;   // 100000 nodes
  const int E = in_sizes[1] / 2;    // 1000000 edges
  const int* src = eidx;
  const int* dst = eidx + E;

  // Workspace carve-out: [h N*64][h1 N*16][sum1 N*64][sum2 N*16][deg N]
  float* h    = (float*)d_ws;
  float* h1   = h    + (size_t)N * 64;
  float* sum1 = h1   + (size_t)N * 16;
  float* sum2 = sum1 + (size_t)N * 64;
  float* deg  = sum2 + (size_t)N * 16;

  // Zero accumulators (sum1|sum2|deg are contiguous) — capture-safe async memset
  hipMemsetAsync(sum1, 0, (size_t)N * (64 + 16 + 1) * sizeof(float), stream);

  const int mtiles = (N + 15) / 16;

  // 1) input projection (WMMA fp32)
  lin_proj_wmma<<<mtiles, 128, 0, stream>>>(x, W_lin, b_lin, h, N);

  // 2) degrees
  degree_kernel<<<(E + 255) / 256, 256, 0, stream>>>(dst, deg, E);

  // 3) layer-1 scatter: 16 float4-chunks per edge
  {
    const long long total = (long long)E * 16;
    scatter_add<<<(int)((total + 255) / 256), 256, 0, stream>>>(
        src, dst, h, sum1, E, /*chunk_shift=*/4, /*F=*/64);
  }

  // 4) SAGE layer 1 fused mean+GEMM+tanh (WMMA fp32)
  combine1_wmma<<<(mtiles + 7) / 8, 256, 0, stream>>>(
      sum1, deg, h, W1_agg, W1_root, b1, h1, N);

  // 5) layer-2 scatter: 4 float4-chunks per edge
  {
    const long long total = (long long)E * 4;
    scatter_add<<<(int)((total + 255) / 256), 256, 0, stream>>>(
        src, dst, h1, sum2, E, /*chunk_shift=*/2, /*F=*/16);
  }

  // 6) SAGE layer 2 fused mean+GEMM+tanh (WMMA fp32)
  combine2_wmma<<<mtiles, 128, 0, stream>>>(
      sum2, deg, h1, W2_agg, W2_root, b2, out, N);
}